// LatentAttentionPooling_25168508354998
// MI455X (gfx1250) — compile-verified
//
#include <hip/hip_runtime.h>
#include <math.h>

// Problem constants (match reference)
constexpr int kH    = 2048;   // hidden
constexpr int kS    = 4096;   // sequence
constexpr int kB    = 16;     // batch
constexpr int kPool = 4096;   // output dim
constexpr int kCH   = 64;     // S-chunks per batch
constexpr int kRows = kS / kCH;  // 64 rows per chunk
constexpr float kEps = 1e-6f;
constexpr float kRsqrtH = 0.022097086912079612f; // 1/sqrt(2048)

typedef float v2f __attribute__((ext_vector_type(2)));
typedef float v8f __attribute__((ext_vector_type(8)));

__device__ __forceinline__ float wave_sum(float v) {
#pragma unroll
  for (int off = 16; off >= 1; off >>= 1) v += __shfl_xor(v, off, 32);
  return v;
}
__device__ __forceinline__ float wave_max(float v) {
#pragma unroll
  for (int off = 16; off >= 1; off >>= 1) v = fmaxf(v, __shfl_xor(v, off, 32));
  return v;
}

// qv[d] = sum_h latent_query[h] * Wq[d,h]    (one wave per output row)
__global__ __launch_bounds__(256) void lap_qv_kernel(
    const float* __restrict__ Wq, const float* __restrict__ lq,
    float* __restrict__ qv) {
  int w = threadIdx.x >> 5, lane = threadIdx.x & 31;
  int d = blockIdx.x * 8 + w;
  const float4* wr = (const float4*)(Wq + (size_t)d * kH);
  const float4* q4 = (const float4*)lq;
  float acc = 0.f;
#pragma unroll
  for (int i = 0; i < 16; ++i) {
    float4 a = wr[i * 32 + lane];
    float4 b = q4[i * 32 + lane];
    acc += a.x * b.x + a.y * b.y + a.z * b.z + a.w * b.w;
  }
  acc = wave_sum(acc);
  if (lane == 0) qv[d] = acc;
}

// k_eff[h] = (sum_d qv[d] * Wk[d,h]) / sqrt(H)   (thread h, coalesced over h)
__global__ __launch_bounds__(256) void lap_keff_kernel(
    const float* __restrict__ Wk, const float* __restrict__ qv,
    float* __restrict__ keff) {
  int h = blockIdx.x * 256 + threadIdx.x;
  float acc = 0.f;
  for (int d = 0; d < kH; ++d) acc += qv[d] * Wk[(size_t)d * kH + h];
  keff[h] = acc * kRsqrtH;
}

// Flash-style chunk: scores -> local softmax stats -> weighted row-sum of X.
// grid (kCH, kB), 256 threads. Outputs per chunk: acc[H], (m, l).
__global__ __launch_bounds__(256) void lap_pool_chunk_kernel(
    const float* __restrict__ X, const int* __restrict__ mask,
    const float* __restrict__ keff, float* __restrict__ acc_out,
    float* __restrict__ ml_out) {
  __shared__ float lds_keff[kH];
  __shared__ float lds_sc[kRows];
  __shared__ float lds_w[kRows];
  const int b = blockIdx.y, chunk = blockIdx.x;
  const int s0 = chunk * kRows;
  const int t = threadIdx.x;
  float4* k4 = (float4*)lds_keff;
  const float4* kg4 = (const float4*)keff;
  k4[t] = kg4[t];
  k4[256 + t] = kg4[256 + t];
  __syncthreads();

  const int w = t >> 5, lane = t & 31;
  const float* xb = X + ((size_t)b * kS + s0) * kH;

  // Phase A: one wave per row, dot(X_row, k_eff) via shuffle reduce
#pragma unroll
  for (int r = 0; r < kRows / 8; ++r) {
    int sl = w * (kRows / 8) + r;
    const float4* xr = (const float4*)(xb + (size_t)sl * kH);
    float acc = 0.f;
#pragma unroll 4
    for (int i = 0; i < 16; ++i) {
      float4 x = xr[i * 32 + lane];
      float4 kk = k4[i * 32 + lane];
      acc += x.x * kk.x + x.y * kk.y + x.z * kk.z + x.w * kk.w;
    }
    acc = wave_sum(acc);
    if (lane == 0) {
      int s = s0 + sl;
      lds_sc[sl] = (mask[(size_t)b * kS + s] == 0) ? -INFINITY : acc;
    }
  }
  __syncthreads();

  // wave 0: chunk max, unnormalized exp weights, partial denominator
  if (w == 0) {
    float a0 = lds_sc[lane], a1 = lds_sc[lane + 32];
    float mx = wave_max(fmaxf(a0, a1));
    float e0 = (a0 == -INFINITY) ? 0.f : __expf(a0 - mx);
    float e1 = (a1 == -INFINITY) ? 0.f : __expf(a1 - mx);
    lds_w[lane] = e0;
    lds_w[lane + 32] = e1;
    float l = wave_sum(e0 + e1);
    if (lane == 0) {
      ml_out[((size_t)b * kCH + chunk) * 2 + 0] = mx;
      ml_out[((size_t)b * kCH + chunk) * 2 + 1] = l;
    }
  }
  __syncthreads();

  // Phase B: acc[h] += w_s * X[b,s,h]  (chunk is L2-hot from Phase A)
  float ax0 = 0.f, ay0 = 0.f, az0 = 0.f, aw0 = 0.f;
  float ax1 = 0.f, ay1 = 0.f, az1 = 0.f, aw1 = 0.f;
  for (int sl = 0; sl < kRows; ++sl) {
    float wt = lds_w[sl];
    const float4* xr = (const float4*)(xb + (size_t)sl * kH);
    float4 x0 = xr[t];
    float4 x1 = xr[256 + t];
    ax0 += wt * x0.x; ay0 += wt * x0.y; az0 += wt * x0.z; aw0 += wt * x0.w;
    ax1 += wt * x1.x; ay1 += wt * x1.y; az1 += wt * x1.z; aw1 += wt * x1.w;
  }
  float4* d4 = (float4*)(acc_out + ((size_t)b * kCH + chunk) * kH);
  d4[t] = make_float4(ax0, ay0, az0, aw0);
  d4[256 + t] = make_float4(ax1, ay1, az1, aw1);
}

// Merge chunk partials: xbar[b,h] = sum_c exp(m_c-M) acc_c[h] / (sum_c l_c exp(m_c-M))
__global__ __launch_bounds__(256) void lap_combine_kernel(
    const float* __restrict__ acc_in, const float* __restrict__ ml_in,
    float* __restrict__ xbar) {
  __shared__ float lm[kCH], ll[kCH], lf[kCH];
  __shared__ float s_invL;
  const int b = blockIdx.x, t = threadIdx.x;
  if (t < kCH) {
    lm[t] = ml_in[((size_t)b * kCH + t) * 2 + 0];
    ll[t] = ml_in[((size_t)b * kCH + t) * 2 + 1];
  }
  __syncthreads();
  if (t == 0) {
    float M = -INFINITY;
    for (int c = 0; c < kCH; ++c) M = fmaxf(M, lm[c]);
    float L = 0.f;
    for (int c = 0; c < kCH; ++c) {
      float f = (lm[c] == -INFINITY) ? 0.f : __expf(lm[c] - M);
      lf[c] = f;
      L += ll[c] * f;
    }
    s_invL = (L > 0.f) ? 1.f / L : 0.f;
  }
  __syncthreads();
  float ax0 = 0.f, ay0 = 0.f, az0 = 0.f, aw0 = 0.f;
  float ax1 = 0.f, ay1 = 0.f, az1 = 0.f, aw1 = 0.f;
  for (int c = 0; c < kCH; ++c) {
    float f = lf[c];
    const float4* a4 = (const float4*)(acc_in + ((size_t)b * kCH + c) * kH);
    float4 x0 = a4[t];
    float4 x1 = a4[256 + t];
    ax0 += f * x0.x; ay0 += f * x0.y; az0 += f * x0.z; aw0 += f * x0.w;
    ax1 += f * x1.x; ay1 += f * x1.y; az1 += f * x1.z; aw1 += f * x1.w;
  }
  float inv = s_invL;
  float4* o4 = (float4*)(xbar + (size_t)b * kH);
  o4[t] = make_float4(ax0 * inv, ay0 * inv, az0 * inv, aw0 * inv);
  o4[256 + t] = make_float4(ax1 * inv, ay1 * inv, az1 * inv, aw1 * inv);
}

// RMS-norm pooled row + fold norm_w: anorm[b,h] = pooled[b,h]*rsqrt(mean(p^2)+eps)*norm_w[h]
__global__ __launch_bounds__(256) void lap_rmsnorm_kernel(
    const float* __restrict__ pooled, const float* __restrict__ norm_w,
    float* __restrict__ anorm) {
  __shared__ float lds_p[8];
  __shared__ float s_inv;
  const int b = blockIdx.x, t = threadIdx.x;
  const float4* p4 = (const float4*)(pooled + (size_t)b * kH);
  float4 x0 = p4[t], x1 = p4[256 + t];
  float ss = x0.x * x0.x + x0.y * x0.y + x0.z * x0.z + x0.w * x0.w +
             x1.x * x1.x + x1.y * x1.y + x1.z * x1.z + x1.w * x1.w;
  ss = wave_sum(ss);
  int w = t >> 5, lane = t & 31;
  if (lane == 0) lds_p[w] = ss;
  __syncthreads();
  if (t == 0) {
    float tot = 0.f;
    for (int i = 0; i < 8; ++i) tot += lds_p[i];
    s_inv = rsqrtf(tot / (float)kH + kEps);
  }
  __syncthreads();
  float inv = s_inv;
  const float4* w4 = (const float4*)norm_w;
  float4 g0 = w4[t], g1 = w4[256 + t];
  float4* o4 = (float4*)(anorm + (size_t)b * kH);
  o4[t] = make_float4(x0.x * inv * g0.x, x0.y * inv * g0.y,
                      x0.z * inv * g0.z, x0.w * inv * g0.w);
  o4[256 + t] = make_float4(x1.x * inv * g1.x, x1.y * inv * g1.y,
                            x1.z * inv * g1.z, x1.w * inv * g1.w);
}

// D[16,N] = A[16,kH] @ W[N,kH]^T via V_WMMA_F32_16X16X4_F32.
// One wave per 16-column N-tile; 512 accumulating WMMAs over K=2048.
// A fragment (16x4 f32): lanes 0-15 -> M rows, K pair {k0+0,k0+1};
// lanes 16-31 -> same M rows, K pair {k0+2,k0+3}. B mirrors with N rows.
__global__ __launch_bounds__(256) void lap_gemm16_wmma_kernel(
    const float* __restrict__ A, const float* __restrict__ W,
    float* __restrict__ D, int N) {
  const int w = threadIdx.x >> 5, lane = threadIdx.x & 31;
  const int ntile = blockIdx.x * 8 + w;
  const int m = lane & 15;
  const int koff = (lane < 16) ? 0 : 2;
  const int n = ntile * 16 + m;
  const float* arow = A + (size_t)m * kH + koff;
  const float* wrow = W + (size_t)n * kH + koff;
  v8f c = {0.f, 0.f, 0.f, 0.f, 0.f, 0.f, 0.f, 0.f};
#pragma unroll 8
  for (int k0 = 0; k0 < kH; k0 += 4) {
    v2f a  = *(const v2f*)(arow + k0);
    v2f bv = *(const v2f*)(wrow + k0);
    c = __builtin_amdgcn_wmma_f32_16x16x4_f32(
        /*neg_a=*/false, a, /*neg_b=*/false, bv,
        /*c_mod=*/(short)0, c, /*reuse_a=*/false, /*reuse_b=*/false);
  }
  // C/D layout: VGPR r, lanes 0-15 -> (M=r, N=lane); lanes 16-31 -> (M=8+r, N=lane-16)
  const int row0 = (lane < 16) ? 0 : 8;
#pragma unroll
  for (int r = 0; r < 8; ++r) D[(size_t)(row0 + r) * N + n] = c[r];
}

extern "C" void kernel_launch(void* const* d_in, const int* in_sizes, int n_in,
                              void* d_out, int out_size, void* d_ws, size_t ws_size,
                              hipStream_t stream) {
  (void)in_sizes; (void)n_in; (void)out_size; (void)ws_size;
  const float* X    = (const float*)d_in[0];  // [B,S,H]
  const int*   mask = (const int*)d_in[1];    // [B,S]
  const float* lq   = (const float*)d_in[2];  // [1,1,H]
  const float* Wq   = (const float*)d_in[3];  // [H,H]
  const float* Wk   = (const float*)d_in[4];  // [H,H]
  const float* Wv   = (const float*)d_in[5];  // [H,H]
  const float* Wo   = (const float*)d_in[6];  // [POOL,H]
  const float* nw   = (const float*)d_in[7];  // [H]
  float* out = (float*)d_out;                 // [B,POOL]
  float* ws  = (float*)d_ws;

  float* qv     = ws;                               // kH
  float* keff   = qv + kH;                          // kH
  float* cacc   = keff + kH;                        // kB*kCH*kH
  float* cml    = cacc + (size_t)kB * kCH * kH;     // kB*kCH*2
  float* xbar   = cml + (size_t)kB * kCH * 2;       // kB*kH
  float* pooled = xbar + (size_t)kB * kH;           // kB*kH
  float* anorm  = pooled + (size_t)kB * kH;         // kB*kH

  lap_qv_kernel<<<kH / 8, 256, 0, stream>>>(Wq, lq, qv);
  lap_keff_kernel<<<kH / 256, 256, 0, stream>>>(Wk, qv, keff);
  lap_pool_chunk_kernel<<<dim3(kCH, kB), 256, 0, stream>>>(X, mask, keff, cacc, cml);
  lap_combine_kernel<<<kB, 256, 0, stream>>>(cacc, cml, xbar);
  lap_gemm16_wmma_kernel<<<kH / 16 / 8, 256, 0, stream>>>(xbar, Wv, pooled, kH);
  lap_rmsnorm_kernel<<<kB, 256, 0, stream>>>(pooled, nw, anorm);
  lap_gemm16_wmma_kernel<<<kPool / 16 / 8, 256, 0, stream>>>(anorm, Wo, out, kPool);
}